// GNN_87522843558077
// MI455X (gfx1250) — compile-verified
//
#include <hip/hip_runtime.h>
#include <hip/hip_bf16.h>

// GraphSAGE 4-layer forward for MI455X (gfx1250, wave32).
// Aggregation: wave-per-edge gather + fp32 atomic scatter (L2-resident).
// GEMM: V_WMMA_F32_16X16X4_F32, 16x16 output tile per wave, K=256 fused
// (mean @ Wl^T then h @ Wr^T into one fp32 accumulator).

typedef __attribute__((ext_vector_type(2))) float v2f;
typedef __attribute__((ext_vector_type(8))) float v8f;

#define DFEAT 128
#define APITCH 132   // LDS row pitch in floats: 528B, 16B-aligned, breaks 64-bank periodicity

// ---------------------------------------------------------------- utilities
__global__ void gnn_zero_f32(float* __restrict__ p, int n) {
    int i = blockIdx.x * blockDim.x + threadIdx.x;
    int stride = gridDim.x * blockDim.x;
    for (; i < n; i += stride) p[i] = 0.0f;
}

__global__ void gnn_count_edges(const int* __restrict__ ei, unsigned* __restrict__ cnt, int E) {
    int e = blockIdx.x * blockDim.x + threadIdx.x;
    if (e < E) atomicAdd(&cnt[ei[E + e]], 1u);   // dst row of edge_index
}

__global__ void gnn_inv_counts(const unsigned* __restrict__ cnt, float* __restrict__ inv, int N) {
    int n = blockIdx.x * blockDim.x + threadIdx.x;
    if (n < N) inv[n] = 1.0f / fmaxf((float)cnt[n], 1.0f);
}

// --------------------------------------------------- edge gather + scatter
// One wave32 per edge: lane L moves features [4L, 4L+4) (float4 = coalesced
// 512B gather per wave), then 4 global_atomic_add_f32 into summed[dst].
__global__ void gnn_scatter_edges(const float* __restrict__ h, const int* __restrict__ ei,
                                  float* __restrict__ summed, int E) {
    int gid  = blockIdx.x * blockDim.x + threadIdx.x;
    int wave = gid >> 5;
    int lane = gid & 31;
    if (wave >= E) return;
    int src = ei[wave];
    int dst = ei[E + wave];
    const float4 v = *(const float4*)(h + (size_t)src * DFEAT + lane * 4);
    float* o = summed + (size_t)dst * DFEAT + lane * 4;
    atomicAdd(o + 0, v.x);
    atomicAdd(o + 1, v.y);
    atomicAdd(o + 2, v.z);
    atomicAdd(o + 3, v.w);
}

// --------------------------------------------------------------- SAGE GEMM
// out[n, j] = bl[j] + sum_k mean[n,k]*Wl[j,k] + sum_k h[n,k]*Wr[j,k]
// Block: 256 threads = 8 waves; one 16-row tile per block; wave w owns
// output columns [16w, 16w+16). A tiles staged in LDS; B read from global
// (Wl/Wr are 64 KB each -> WGP$/L2 resident).
__global__ __launch_bounds__(256)
void gnn_sage_gemm(const float* __restrict__ summed, const float* __restrict__ inv,
                   const float* __restrict__ h,
                   const float* __restrict__ Wl, const float* __restrict__ bl,
                   const float* __restrict__ Wr,
                   float* __restrict__ out, int N, int relu) {
    __shared__ float a_mean[16 * APITCH];
    __shared__ float a_h[16 * APITCH];

    const int r0 = blockIdx.x * 16;
    const int t  = threadIdx.x;

    // Stage A tiles (coalesced float4 global loads; mean scaled by inv_cnt).
    for (int i = t; i < 16 * (DFEAT / 4); i += 256) {
        int row  = i >> 5;            // 32 float4 per row
        int c4   = (i & 31) * 4;
        int grow = r0 + row;
        float4 m4 = make_float4(0.f, 0.f, 0.f, 0.f);
        float4 h4 = make_float4(0.f, 0.f, 0.f, 0.f);
        if (grow < N) {
            m4 = *(const float4*)(summed + (size_t)grow * DFEAT + c4);
            h4 = *(const float4*)(h + (size_t)grow * DFEAT + c4);
            float s = inv[grow];
            m4.x *= s; m4.y *= s; m4.z *= s; m4.w *= s;
        }
        *(float4*)&a_mean[row * APITCH + c4] = m4;
        *(float4*)&a_h[row * APITCH + c4]   = h4;
    }
    __syncthreads();

    // WMMA fragment indexing (ISA 7.12.2, 32-bit 16x4 A / 4x16 B / 16x16 C):
    //   lane L: m/n = L&15, kk = L>>4; A/B VGPR p holds K = kbase + p + 2*kk;
    //   C/D VGPR r holds row M = r + 8*kk.
    const int wave = t >> 5;
    const int lane = t & 31;
    const int kk   = lane >> 4;
    const int mn   = lane & 15;
    const int ncol = wave * 16 + mn;

    const float* wlrow = Wl + (size_t)ncol * DFEAT;  // Wl[j,k] row == B[k][j] column data
    const float* wrrow = Wr + (size_t)ncol * DFEAT;

    v8f acc = {};
    #pragma unroll 8
    for (int ks = 0; ks < 32; ++ks) {           // mean @ Wl^T, K = 128
        int kb = ks * 4 + kk * 2;
        v2f a = *(const v2f*)&a_mean[mn * APITCH + kb];
        v2f b = *(const v2f*)(wlrow + kb);
        acc = __builtin_amdgcn_wmma_f32_16x16x4_f32(false, a, false, b,
                                                    (short)0, acc, false, false);
    }
    #pragma unroll 8
    for (int ks = 0; ks < 32; ++ks) {           // h @ Wr^T, K = 128
        int kb = ks * 4 + kk * 2;
        v2f a = *(const v2f*)&a_h[mn * APITCH + kb];
        v2f b = *(const v2f*)(wrrow + kb);
        acc = __builtin_amdgcn_wmma_f32_16x16x4_f32(false, a, false, b,
                                                    (short)0, acc, false, false);
    }

    const float bias = bl[ncol];
    #pragma unroll
    for (int r = 0; r < 8; ++r) {
        int grow = r0 + r + kk * 8;
        float v = acc[r] + bias;
        if (relu) v = fmaxf(v, 0.0f);
        if (grow < N) out[(size_t)grow * DFEAT + ncol] = v;
    }
}

// ------------------------------------------------------------------ driver
extern "C" void kernel_launch(void* const* d_in, const int* in_sizes, int n_in,
                              void* d_out, int out_size, void* d_ws, size_t ws_size,
                              hipStream_t stream) {
    const float* x  = (const float*)d_in[0];
    const int*   ei = (const int*)d_in[1];
    const int N = in_sizes[0] / DFEAT;   // 50000
    const int E = in_sizes[1] / 2;       // 800000

    // Workspace layout
    char* ws = (char*)d_ws;
    size_t off = 0;
    auto carve = [&](size_t bytes) -> void* {
        void* p = ws + off;
        off += (bytes + 255) & ~(size_t)255;
        return p;
    };
    float*    summed = (float*)carve((size_t)N * DFEAT * sizeof(float));
    float*    h0     = (float*)carve((size_t)N * DFEAT * sizeof(float));
    float*    h1     = (float*)carve((size_t)N * DFEAT * sizeof(float));
    unsigned* cnt    = (unsigned*)carve((size_t)N * sizeof(unsigned));
    float*    inv    = (float*)carve((size_t)N * sizeof(float));

    // Degree counts depend only on edge_index: compute once, reuse 4 layers.
    gnn_zero_f32<<<256, 256, 0, stream>>>((float*)cnt, N);
    gnn_count_edges<<<(E + 255) / 256, 256, 0, stream>>>(ei, cnt, E);
    gnn_inv_counts<<<(N + 255) / 256, 256, 0, stream>>>(cnt, inv, N);

    const float* hin = x;
    float* outs[4] = {h0, h1, h0, (float*)d_out};
    const int nfeat = N * DFEAT;
    const int scatter_blocks = (int)(((size_t)E * 32 + 255) / 256);

    for (int l = 0; l < 4; ++l) {
        const float* Wl = (const float*)d_in[2 + 3 * l];
        const float* bl = (const float*)d_in[3 + 3 * l];
        const float* Wr = (const float*)d_in[4 + 3 * l];

        gnn_zero_f32<<<2048, 256, 0, stream>>>(summed, nfeat);
        gnn_scatter_edges<<<scatter_blocks, 256, 0, stream>>>(hin, ei, summed, E);
        gnn_sage_gemm<<<(N + 15) / 16, 256, 0, stream>>>(summed, inv, hin,
                                                         Wl, bl, Wr, outs[l], N,
                                                         (l == 0) ? 1 : 0);
        hin = outs[l];
    }
}